// GNNEncoder_19859928777344
// MI455X (gfx1250) — compile-verified
//
#include <hip/hip_runtime.h>
#include <cstddef>

typedef __attribute__((ext_vector_type(2))) float v2f;
typedef __attribute__((ext_vector_type(8))) float v8f;

namespace {

constexpr int kN = 50000;   // nodes
constexpr int kE = 800000;  // edges
constexpr int kD = 128;     // feature width of every scatter input

// ---------------------------------------------------------------- utilities
__global__ void zero_f32(float* __restrict__ p, int n) {
  int i = blockIdx.x * blockDim.x + threadIdx.x;
  if (i < n) p[i] = 0.0f;
}

__global__ void vec_add(const float* __restrict__ a, const float* __restrict__ b,
                        float* __restrict__ o, int n) {
  int i = blockIdx.x * blockDim.x + threadIdx.x;
  if (i < n) o[i] = a[i] + b[i];
}

// in-degree count (as float so we can atomicAdd and invert in place)
__global__ void count_dst(const long long* __restrict__ ei, float* __restrict__ cnt) {
  int e = blockIdx.x * blockDim.x + threadIdx.x;
  if (e < kE) atomicAdd(&cnt[(int)ei[kE + e]], 1.0f);
}

__global__ void recip_clamp(float* __restrict__ c, int n) {
  int i = blockIdx.x * blockDim.x + threadIdx.x;
  if (i < n) c[i] = 1.0f / fmaxf(c[i], 1.0f);
}

// one wave32 per edge: 128 features = 4 coalesced f32 atomics per lane
__global__ void scatter_sum(const float* __restrict__ x,
                            const long long* __restrict__ ei,
                            float* __restrict__ agg) {
  int t = blockIdx.x * blockDim.x + threadIdx.x;
  int e = t >> 5;
  if (e >= kE) return;
  int lane = t & 31;
  int s = (int)ei[e];        // src row
  int d = (int)ei[kE + e];   // dst row
  const float* __restrict__ xr = x + (size_t)s * kD;
  float* __restrict__ ar = agg + (size_t)d * kD;
#pragma unroll
  for (int k = 0; k < kD; k += 32)
    atomicAdd(&ar[k + lane], xr[k + lane]);
}

// ---------------------------------------------------------------- fused GEMM
// out[m,n] = relu?( (agg[m,:]*inv[m]) @ Wl + xin[m,:] @ Wrs + bias[n] )
// One wave per 16x16 tile. fp32 WMMA (16x16x4) keeps exact fp32 math; the
// GEMM is far from the bottleneck so there is no reason to downcast.
template <int DOUT, bool RELU>
__global__ void __launch_bounds__(256) sage_gemm(
    const float* __restrict__ agg, const float* __restrict__ xin,
    const float* __restrict__ inv, const float* __restrict__ Wl,
    const float* __restrict__ Wrs, const float* __restrict__ bias,
    float* __restrict__ out) {
  constexpr int COLT = DOUT / 16;
  constexpr int TILES = (kN / 16) * COLT;
  int wave = (int)((blockIdx.x * blockDim.x + threadIdx.x) >> 5);
  if (wave >= TILES) return;  // wave-uniform: EXEC stays all-ones for WMMA

  int rowTile = wave / COLT;
  int colTile = wave % COLT;
  int row0 = rowTile * 16;
  int col0 = colTile * 16;
  int lane = threadIdx.x & 31;
  int hi = lane >> 4;   // half-wave selects K pair (A) / K row pair (B)
  int lm = lane & 15;
  int m = row0 + lm;    // A row owned by this lane
  int n = col0 + lm;    // B/C column owned by this lane

  float scale = inv[m];  // fold mean normalization into the A fragment
  const float* __restrict__ aRow = agg + (size_t)m * kD;
  const float* __restrict__ xRow = xin + (size_t)m * kD;

  v8f c = {};
#pragma unroll
  for (int k0 = 0; k0 < kD; k0 += 4) {
    int ka = k0 + (hi ? 2 : 0);
    // A fragments (ISA 16x4 layout: lanes 0-15 -> K=k0..k0+1, 16-31 -> +2)
    v2f aA = *(const v2f*)(aRow + ka);
    aA[0] *= scale;
    aA[1] *= scale;
    v2f aX = *(const v2f*)(xRow + ka);
    // B fragments (4x16, row K striped across a half-wave per VGPR)
    v2f bL, bR;
    bL[0] = Wl[(size_t)ka * DOUT + n];
    bL[1] = Wl[(size_t)(ka + 1) * DOUT + n];
    bR[0] = Wrs[(size_t)ka * DOUT + n];
    bR[1] = Wrs[(size_t)(ka + 1) * DOUT + n];
    c = __builtin_amdgcn_wmma_f32_16x16x4_f32(false, aA, false, bL,
                                              (short)0, c, false, false);
    c = __builtin_amdgcn_wmma_f32_16x16x4_f32(false, aX, false, bR,
                                              (short)0, c, false, false);
  }

  float b = bias[n];
#pragma unroll
  for (int v = 0; v < 8; v++) {  // C/D layout: M = row0 + v + 8*hi, N = n
    int mm = row0 + v + (hi ? 8 : 0);
    float val = c[v] + b;
    if (RELU) val = fmaxf(val, 0.0f);
    out[(size_t)mm * DOUT + n] = val;
  }
}

}  // namespace

// ---------------------------------------------------------------- launcher
extern "C" void kernel_launch(void* const* d_in, const int* in_sizes, int n_in,
                              void* d_out, int out_size, void* d_ws, size_t ws_size,
                              hipStream_t stream) {
  const float* x = (const float*)d_in[0];
  const long long* ei = (const long long*)d_in[1];  // int64 [2,E]
  const float* Wl0 = (const float*)d_in[4];
  const float* bl0 = (const float*)d_in[5];
  const float* Wr0 = (const float*)d_in[6];
  const float* Ws0 = (const float*)d_in[7];
  const float* bs0 = (const float*)d_in[8];
  const float* Wl1 = (const float*)d_in[9];
  const float* bl1 = (const float*)d_in[10];
  const float* Wr1 = (const float*)d_in[11];
  const float* Ws1 = (const float*)d_in[12];
  const float* bs1 = (const float*)d_in[13];
  const float* Wl2 = (const float*)d_in[14];
  const float* bl2 = (const float*)d_in[15];
  const float* Wr2 = (const float*)d_in[16];
  const float* Ws2 = (const float*)d_in[17];
  const float* bs2 = (const float*)d_in[18];
  float* out = (float*)d_out;

  // workspace layout (floats)
  float* ws = (float*)d_ws;
  size_t off = 0;
  float* agg = ws + off;  off += (size_t)kN * kD;   // 25.6 MB
  float* h0  = ws + off;  off += (size_t)kN * kD;   // 25.6 MB
  float* h1  = ws + off;  off += (size_t)kN * kD;   // 25.6 MB
  float* inv = ws + off;  off += (size_t)kN;        // deg^-1
  float* Wrs0 = ws + off; off += 128 * 128;
  float* Wrs1 = ws + off; off += 128 * 128;
  float* Wrs2 = ws + off; off += 128 * 64;
  float* bsum0 = ws + off; off += 128;
  float* bsum1 = ws + off; off += 128;
  float* bsum2 = ws + off; off += 64;

  const int T = 256;
  auto blocks = [](long long n, int t) { return (int)((n + t - 1) / t); };

  // fold Wr+Ws and bl+bs (tiny)
  vec_add<<<blocks(128 * 128, T), T, 0, stream>>>(Wr0, Ws0, Wrs0, 128 * 128);
  vec_add<<<blocks(128 * 128, T), T, 0, stream>>>(Wr1, Ws1, Wrs1, 128 * 128);
  vec_add<<<blocks(128 * 64, T), T, 0, stream>>>(Wr2, Ws2, Wrs2, 128 * 64);
  vec_add<<<1, T, 0, stream>>>(bl0, bs0, bsum0, 128);
  vec_add<<<1, T, 0, stream>>>(bl1, bs1, bsum1, 128);
  vec_add<<<1, T, 0, stream>>>(bl2, bs2, bsum2, 64);

  // degree -> 1/max(deg,1), shared by all layers
  zero_f32<<<blocks(kN, T), T, 0, stream>>>(inv, kN);
  count_dst<<<blocks(kE, T), T, 0, stream>>>(ei, inv);
  recip_clamp<<<blocks(kN, T), T, 0, stream>>>(inv, kN);

  const long long scatterThreads = (long long)kE * 32;
  const int featN = kN * kD;

  // ---- layer 0: h0 = relu(mean(x[src]->dst) @ Wl0 + x @ (Wr0+Ws0) + b)
  zero_f32<<<blocks(featN, T), T, 0, stream>>>(agg, featN);
  scatter_sum<<<blocks(scatterThreads, T), T, 0, stream>>>(x, ei, agg);
  sage_gemm<128, true><<<blocks((kN / 16) * 8 * 32, T), T, 0, stream>>>(
      agg, x, inv, Wl0, Wrs0, bsum0, h0);

  // ---- layer 1
  zero_f32<<<blocks(featN, T), T, 0, stream>>>(agg, featN);
  scatter_sum<<<blocks(scatterThreads, T), T, 0, stream>>>(h0, ei, agg);
  sage_gemm<128, true><<<blocks((kN / 16) * 8 * 32, T), T, 0, stream>>>(
      agg, h0, inv, Wl1, Wrs1, bsum1, h1);

  // ---- layer 2 (no relu, DOUT=64) -> d_out
  zero_f32<<<blocks(featN, T), T, 0, stream>>>(agg, featN);
  scatter_sum<<<blocks(scatterThreads, T), T, 0, stream>>>(h1, ei, agg);
  sage_gemm<64, false><<<blocks((kN / 16) * 4 * 32, T), T, 0, stream>>>(
      agg, h1, inv, Wl2, Wrs2, bsum2, out);
}